// SmoothnessLoss_33921651703905
// MI455X (gfx1250) — compile-verified
//
#include <hip/hip_runtime.h>
#include <hip/hip_bf16.h>
#include <stdint.h>

// ---------------- constants ----------------
#define BB   16
#define HH   1024
#define WW   1024
#define TILE_W 128           // interior tile width  (== TDM row width, power of two for pad_interval)
#define TILE_H 16            // interior tile height
#define ROWS   18            // TILE_H + 2 (vertical halo)
#define PITCH  132           // LDS row pitch in floats (128 data + 4 pad from TDM pad_amount)
#define NT     8             // tiles per strip (strip = 128 rows)
#define RP     (ROWS * PITCH)        // floats per LDS tile buffer
#define NBLK   (BB * 8 * 8)          // 16 batches * 8 x-tiles * 8 y-strips = 1024
#define INV_N  (1.0f / 67108864.0f)  // 1 / (16*4*1024*1024)

typedef __attribute__((ext_vector_type(4))) unsigned int v4u;
typedef __attribute__((ext_vector_type(8))) int          v8i;
typedef __attribute__((ext_vector_type(4))) int          v4i;
typedef __attribute__((ext_vector_type(2))) float        v2f;
typedef __attribute__((ext_vector_type(8))) float        v8f;

// ---------------- TDM: DMA one 2D tile (<=18 rows x 128 cols, fp32) into LDS ----------------
// D# per CDNA5 ISA ch.8 (CDNA5 ISA):
//   group0: [count=1][lds_addr][global_addr 57b | type=2]
//   group1: data_size=4B, pad_enable, pad_interval=128 DW, pad_amount=4 DW -> LDS pitch 132 DW,
//           tensor_dim0/1 = remaining extent from tile start (right/bottom OOB reads return zero),
//           tile_dim0=128, tile_dim1=18-skipY, dim0_stride=1024.
__device__ __forceinline__ void tdm_load_tile(uint32_t lds_byte_addr,
                                              const float* gbase,
                                              int batch, int y0, int x0) {
  int startY = y0 - 1;                 // include top halo row
  uint32_t skipY = 0u;
  if (startY < 0) { skipY = 1u; startY = 0; }   // y==0: top halo pre-zeroed in LDS instead
  uint64_t gaddr = (uint64_t)(uintptr_t)gbase +
                   ((((uint64_t)batch * HH) + (uint64_t)startY) * WW + (uint64_t)x0) * 4ull;
  uint32_t ldsA = lds_byte_addr + skipY * (uint32_t)(PITCH * 4);
  uint32_t remX = (uint32_t)(WW - x0);       // remaining row extent from tile start
  uint32_t remY = (uint32_t)(HH - startY);   // remaining col extent from tile start

  v4u g0;
  g0.x = 1u;                                  // count=1, user mode, gather off
  g0.y = ldsA;                                // lds_addr (bytes)
  g0.z = (uint32_t)gaddr;                     // global_addr[31:0]
  g0.w = (uint32_t)((gaddr >> 32) & 0x01FFFFFFull) | 0x80000000u;  // addr[56:32] | type=2

  v8i g1;
  g1.s0 = (int)((2u << 16)        // data_size = 4 bytes
              | (1u << 20)        // pad_enable
              | (6u << 22)        // pad_interval: 128 DWORDs per row
              | (3u << 25));      // pad_amount: 4 DWORDs  -> pitch 132
  g1.s1 = (int)((remX & 0xFFFFu) << 16);                               // tensor_dim0 lo16
  g1.s2 = (int)(((remX >> 16) & 0xFFFFu) | ((remY & 0xFFFFu) << 16));  // dim0 hi | dim1 lo
  g1.s3 = (int)(((remY >> 16) & 0xFFFFu) | ((uint32_t)TILE_W << 16));  // dim1 hi | tile_dim0
  g1.s4 = (int)((uint32_t)ROWS - skipY);                               // tile_dim1 | tile_dim2=0
  g1.s5 = (int)WW;                                                     // tensor_dim0_stride lo
  g1.s6 = 0;                                                           // stride hi | dim1_stride lo
  g1.s7 = 0;

  v4i gz; gz.x = 0; gz.y = 0; gz.z = 0; gz.w = 0;
#if defined(__clang_major__) && (__clang_major__ >= 23)
  v8i z8; z8.s0 = 0; z8.s1 = 0; z8.s2 = 0; z8.s3 = 0;
          z8.s4 = 0; z8.s5 = 0; z8.s6 = 0; z8.s7 = 0;
  __builtin_amdgcn_tensor_load_to_lds(g0, g1, gz, gz, z8, 0);
#else
  __builtin_amdgcn_tensor_load_to_lds(g0, g1, gz, gz, 0);
#endif
}

__device__ __forceinline__ float sl1(float v) {
  float a = __builtin_fabsf(v);
  return (a < 1.0f) ? 0.5f * v * v : a - 0.5f;
}

// taps (d = in - tgt) at LDS row rr, cols lx-1, lx, lx+1; edge columns come from global
__device__ __forceinline__ void loadrow(const float* __restrict__ bi, const float* __restrict__ bt,
                                        int rr, int lx, int batch, int y0, int x0,
                                        const float* __restrict__ gin, const float* __restrict__ gtg,
                                        float& t0, float& t1, float& t2) {
  int base = rr * PITCH;
  t1 = bi[base + lx] - bt[base + lx];
  if (lx > 0) {
    t0 = bi[base + lx - 1] - bt[base + lx - 1];
  } else {
    int gy = y0 + rr - 1, gx = x0 - 1;
    if (gx >= 0 && gy >= 0 && gy < HH) {
      size_t idx = ((size_t)batch * HH + (size_t)gy) * WW + (size_t)gx;
      t0 = gin[idx] - gtg[idx];
    } else t0 = 0.0f;
  }
  if (lx < TILE_W - 1) {
    t2 = bi[base + lx + 1] - bt[base + lx + 1];
  } else {
    int gy = y0 + rr - 1, gx = x0 + TILE_W;
    if (gx < WW && gy >= 0 && gy < HH) {
      size_t idx = ((size_t)batch * HH + (size_t)gy) * WW + (size_t)gx;
      t2 = gin[idx] - gtg[idx];
    } else t2 = 0.0f;
  }
}

// ---------------- main kernel: stencil + smooth-L1 + per-block partial sum ----------------
__global__ __launch_bounds__(256) void smoothness_main(const float* __restrict__ gin,
                                                       const float* __restrict__ gtg,
                                                       float* __restrict__ ws) {
  __shared__ float lds[4 * RP];      // [in0][tg0][in1][tg1], 38,016 bytes
  __shared__ float red[8];

  const int tid   = (int)threadIdx.x;
  const int bid   = (int)blockIdx.x;
  const int batch = bid >> 6;
  const int rem   = bid & 63;
  const int xt    = rem >> 3;
  const int ys    = rem & 7;
  const int x0    = xt * TILE_W;
  const int ybase = ys * (NT * TILE_H);
  const int lx    = tid & (TILE_W - 1);
  const int half  = tid >> 7;

  // Pre-zero top halo row of buffer 0 for the y==0 strip (TDM skips row 0 there).
  if (ys == 0 && tid < PITCH) { lds[tid] = 0.0f; lds[RP + tid] = 0.0f; }

  const uint32_t ldsBase = (uint32_t)(uintptr_t)&lds[0];

  if (tid < 32) {  // wave 0 drives the Tensor Data Mover
    tdm_load_tile(ldsBase,            gin, batch, ybase, x0);
    tdm_load_tile(ldsBase + RP * 4u,  gtg, batch, ybase, x0);
  }

  float acc = 0.0f;
  for (int t = 0; t < NT; ++t) {
    const int buf = t & 1;
    if (tid < 32) __builtin_amdgcn_s_wait_tensorcnt(0);
    __syncthreads();                     // tile t visible; everyone done reading buf^1
    if (t + 1 < NT && tid < 32) {        // prefetch tile t+1 into the other buffer
      const uint32_t off = (uint32_t)((buf ^ 1) * 2 * RP * 4);
      tdm_load_tile(ldsBase + off,           gin, batch, ybase + (t + 1) * TILE_H, x0);
      tdm_load_tile(ldsBase + off + RP * 4u, gtg, batch, ybase + (t + 1) * TILE_H, x0);
    }

    const float* bi = lds + (buf ? 2 * RP : 0);
    const float* bt = bi + RP;
    const int y0 = ybase + t * TILE_H;
    const int r0 = half * 8;             // this thread's 8-row run

    float u0, u1, u2, m0, m1, m2, d0, d1, d2;
    loadrow(bi, bt, r0,     lx, batch, y0, x0, gin, gtg, u0, u1, u2);
    loadrow(bi, bt, r0 + 1, lx, batch, y0, x0, gin, gtg, m0, m1, m2);
#pragma unroll
    for (int rr = 0; rr < 8; ++rr) {
      const int r = r0 + rr;
      loadrow(bi, bt, r + 2, lx, batch, y0, x0, gin, gtg, d0, d1, d2);
      const int y = y0 + r;
      const int x = x0 + lx;
      const float c2 = 2.0f * m1;
      const float vx  = m0 - c2 + m2;   // d2/dx2
      const float vy  = u1 - c2 + d1;   // d2/dy2
      const float vd1 = u0 - c2 + d2;   // diag down-right
      const float vd2 = u2 - c2 + d0;   // diag down-left
      float s = 0.0f;
      if (x < WW - 1) s += sl1(vx);                               // mask_x: last col zero
      if (y < HH - 1) s += sl1(vy);                               // mask_y: last row zero
      if (x > 0 && x < WW - 1 && y > 0 && y < HH - 1)             // mask_diag: borders zero
        s += sl1(vd1) + sl1(vd2);
      acc += s;
      u0 = m0; u1 = m1; u2 = m2;
      m0 = d0; m1 = d1; m2 = d2;
    }
  }

  // wave32 tree reduction, then cross-wave via LDS
  for (int off = 16; off; off >>= 1) acc += __shfl_xor(acc, off, 32);
  const int wave = tid >> 5;
  if ((tid & 31) == 0) red[wave] = acc;
  __syncthreads();
  if (tid == 0) {
    float sblk = 0.0f;
#pragma unroll
    for (int w = 0; w < 8; ++w) sblk += red[w];
    ws[bid] = sblk;     // deterministic per-block partial
  }
}

// ---------------- final reduction: 1024 partials -> scalar, via V_WMMA_F32_16X16X4_F32 ----------------
// Lane partials are placed in the A matrix (16x4 f32: vgpr0 = K0|K2, vgpr1 = K1|K3), B = ones(4x16):
// D[m][n] = p_m + p_(m+16); summing the 8 D vgprs per lane + one xor-16 shuffle finishes the sum.
__global__ __launch_bounds__(32) void smoothness_reduce(const float* __restrict__ ws,
                                                        float* __restrict__ out, int n) {
  const int lane = (int)threadIdx.x;
  float s = 0.0f;
  for (int i = lane; i < n; i += 32) s += ws[i];

  v2f a;    a.x = s;    a.y = 0.0f;
  v2f ones; ones.x = 1.0f; ones.y = 1.0f;
  v8f c = {0.0f, 0.0f, 0.0f, 0.0f, 0.0f, 0.0f, 0.0f, 0.0f};
  v8f d = __builtin_amdgcn_wmma_f32_16x16x4_f32(false, a, false, ones, (short)0, c, false, false);

  float r = d.s0 + d.s1 + d.s2 + d.s3 + d.s4 + d.s5 + d.s6 + d.s7;
  r += __shfl_xor(r, 16, 32);
  if (lane == 0) out[0] = r * INV_N;
}

// ---------------- launch ----------------
extern "C" void kernel_launch(void* const* d_in, const int* in_sizes, int n_in,
                              void* d_out, int out_size, void* d_ws, size_t ws_size,
                              hipStream_t stream) {
  const float* input  = (const float*)d_in[0];
  const float* target = (const float*)d_in[1];
  float* out = (float*)d_out;
  float* ws  = (float*)d_ws;   // needs NBLK floats = 4 KB

  smoothness_main<<<NBLK, 256, 0, stream>>>(input, target, ws);
  smoothness_reduce<<<1, 32, 0, stream>>>(ws, out, NBLK);
}